// DEQSequenceModel_27513560498992
// MI455X (gfx1250) — compile-verified
//
#include <hip/hip_runtime.h>

// CDNA5 / gfx1250 DEQ fixed-point kernel.
// Math engine: v_wmma_f32_16x16x4_f32 (exact f32, matches reference precision).
// One wave (32 lanes) owns 16 batch rows; z and inj stay resident on-chip for
// all iterations. Only x (64MB in) and out (64MB out) touch HBM.
//
// Round-2 tuning: inj/decoder GEMMs restructured s-outer/t-inner with partial
// unroll so the prologue's global-load hoisting stays within the 256-VGPR
// budget (kills the scratch spill seen in round 1). Hot loop unchanged: the
// compiler lowers its LDS traffic to ds_*_2addr_* pairs.

typedef __attribute__((ext_vector_type(2))) float v2f;
typedef __attribute__((ext_vector_type(8))) float v8f;

#define HID     128
#define IN_DIM  64
#define OUT_DIM 64
#define WZ_LD   132   // padded LDS row stride (dwords): (132*r + c) % 64 -> conflict-free
#define Z_LD    132
#define WAVES   8     // waves per block

__device__ __forceinline__ v8f relu8(v8f v) {
  v8f r;
#pragma unroll
  for (int i = 0; i < 8; ++i) r[i] = fmaxf(v[i], 0.0f);
  return r;
}

__global__ __launch_bounds__(32 * WAVES, 4) void deq_wmma_f32_kernel(
    const float* __restrict__ x,   const float* __restrict__ Wz,
    const float* __restrict__ Ux,  const float* __restrict__ bvec,
    const float* __restrict__ Wd,  const float* __restrict__ bd,
    const int*   __restrict__ n_iters_p, float* __restrict__ out) {
  extern __shared__ float smem[];
  float* lds_wz = smem;                         // [128][WZ_LD]
  float* lds_z  = smem + HID * WZ_LD;           // [WAVES][16][Z_LD]

  const int tid   = threadIdx.x;
  const int lane  = tid & 31;
  const int wave  = tid >> 5;
  const int mrow  = lane & 15;   // A: row m ; B/C/D: col n
  const int khalf = lane >> 4;   // which K-half of the 4-wide K slab
  const int k2    = 2 * khalf;

  float* zbuf = lds_z + wave * (16 * Z_LD);

  // ---- cooperative stage of Wz into padded LDS (float4 per thread) ----
  {
    const float4* src = (const float4*)Wz;
#pragma unroll 4
    for (int i = tid; i < (HID * HID / 4); i += 32 * WAVES) {
      const int e = i * 4;
      const int r = e >> 7;        // /128
      const int c = e & 127;
      float4 v = src[i];
      float* d = lds_wz + r * WZ_LD + c;
      d[0] = v.x; d[1] = v.y; d[2] = v.z; d[3] = v.w;
    }
  }

  const int n_iters = *n_iters_p;
  const int row0 = (blockIdx.x * WAVES + wave) * 16;

  // ---- x tile in native WMMA A-layout: xa[s] = x[row0+mrow][4s+k2 .. +1] ----
  v2f xa[16];
  {
    const float* xr = x + (size_t)(row0 + mrow) * IN_DIM + k2;
#pragma unroll
    for (int s = 0; s < 16; ++s) xa[s] = *(const v2f*)(xr + 4 * s);
  }

  // ---- inj = x @ Ux^T + b  (8 accumulator tiles, kept in VGPRs) ----
  // s-outer / t-inner, partial unroll: bounded load window, no spills.
  v8f inj[8];
#pragma unroll
  for (int t = 0; t < 8; ++t) {
    const float bb = bvec[16 * t + mrow];
#pragma unroll
    for (int i = 0; i < 8; ++i) inj[t][i] = bb;
  }
  {
    const float* ub = Ux + (size_t)mrow * IN_DIM + k2;
#pragma unroll 4
    for (int s = 0; s < 16; ++s) {
#pragma unroll
      for (int t = 0; t < 8; ++t) {
        v2f bop = *(const v2f*)(ub + (size_t)(16 * t) * IN_DIM + 4 * s);
        inj[t] = __builtin_amdgcn_wmma_f32_16x16x4_f32(
            false, xa[s], false, bop, (short)0, inj[t], false, false);
      }
    }
  }

  __syncthreads();   // Wz resident in LDS

  // ---- fixed-point iterations: z <- relu(z @ Wz^T + inj) ----
  v8f z[8];
  if (n_iters > 0) {
#pragma unroll
    for (int t = 0; t < 8; ++t) z[t] = relu8(inj[t]);   // step 1 (z0 = 0)

    for (int it = 1; it < n_iters; ++it) {
      // z (D layout) -> per-wave LDS scratch; conflict-free b32 stores
#pragma unroll
      for (int t = 0; t < 8; ++t)
#pragma unroll
        for (int r = 0; r < 8; ++r)
          zbuf[(r + 8 * khalf) * Z_LD + 16 * t + mrow] = z[t][r];

      // same-wave LDS ops are in-order: dscnt==0 makes stores visible to loads
      asm volatile("s_wait_dscnt 0" ::: "memory");

      v8f acc[8];
#pragma unroll
      for (int t = 0; t < 8; ++t) acc[t] = inj[t];

#pragma unroll
      for (int s = 0; s < 32; ++s) {
        const v2f a = *(const v2f*)(zbuf + mrow * Z_LD + 4 * s + k2);
#pragma unroll
        for (int t = 0; t < 8; ++t) {
          const v2f bop =
              *(const v2f*)(lds_wz + (16 * t + mrow) * WZ_LD + 4 * s + k2);
          acc[t] = __builtin_amdgcn_wmma_f32_16x16x4_f32(
              false, a, false, bop, (short)0, acc[t], false, false);
        }
      }
#pragma unroll
      for (int t = 0; t < 8; ++t) z[t] = relu8(acc[t]);
    }
  } else {
#pragma unroll
    for (int t = 0; t < 8; ++t)
#pragma unroll
      for (int i = 0; i < 8; ++i) z[t][i] = 0.0f;
  }

  // ---- decoder: out = z @ Wd^T + bd ----
  // route final z through LDS to obtain A-layout operands
#pragma unroll
  for (int t = 0; t < 8; ++t)
#pragma unroll
    for (int r = 0; r < 8; ++r)
      zbuf[(r + 8 * khalf) * Z_LD + 16 * t + mrow] = z[t][r];
  asm volatile("s_wait_dscnt 0" ::: "memory");

  v2f za[32];
#pragma unroll
  for (int s = 0; s < 32; ++s)
    za[s] = *(const v2f*)(zbuf + mrow * Z_LD + 4 * s + k2);

  v8f oc[4];
#pragma unroll
  for (int t = 0; t < 4; ++t) {
    const float bb = bd[16 * t + mrow];
#pragma unroll
    for (int i = 0; i < 8; ++i) oc[t][i] = bb;
  }
  {
    const float* wb = Wd + (size_t)mrow * HID + k2;
#pragma unroll 8
    for (int s = 0; s < 32; ++s) {
#pragma unroll
      for (int t = 0; t < 4; ++t) {
        v2f bop = *(const v2f*)(wb + (size_t)(16 * t) * HID + 4 * s);
        oc[t] = __builtin_amdgcn_wmma_f32_16x16x4_f32(
            false, za[s], false, bop, (short)0, oc[t], false, false);
      }
    }
  }
#pragma unroll
  for (int t = 0; t < 4; ++t)
#pragma unroll
    for (int r = 0; r < 8; ++r)
      out[(size_t)(row0 + r + 8 * khalf) * OUT_DIM + 16 * t + mrow] = oc[t][r];
}

extern "C" void kernel_launch(void* const* d_in, const int* in_sizes, int n_in,
                              void* d_out, int out_size, void* d_ws, size_t ws_size,
                              hipStream_t stream) {
  const float* x    = (const float*)d_in[0];
  const float* Wz   = (const float*)d_in[1];
  const float* Ux   = (const float*)d_in[2];
  const float* bvec = (const float*)d_in[3];
  const float* Wd   = (const float*)d_in[4];
  const float* bd   = (const float*)d_in[5];
  const int*   nit  = (const int*)d_in[6];
  float* out = (float*)d_out;

  const int B = in_sizes[0] / IN_DIM;          // 262144
  const int rows_per_block = 16 * WAVES;       // 128
  const int blocks = B / rows_per_block;       // 2048

  const size_t shmem = (size_t)(HID * WZ_LD + WAVES * 16 * Z_LD) * sizeof(float);

  deq_wmma_f32_kernel<<<blocks, 32 * WAVES, shmem, stream>>>(
      x, Wz, Ux, bvec, Wd, bd, nit, out);
}